// ConditionalSofmax_83726092468743
// MI455X (gfx1250) — compile-verified
//
#include <hip/hip_runtime.h>
#include <math.h>

// Tree: branching [64,16,16]
//   nodes: [0,64) level0 | [64,1088) level1 | [1088,17472) level2
//   groups: group0 = 64 level0 nodes; then 1088 contiguous groups of 16
//           (groups 0..63 = level1 children of parent p; 64..1087 = level2
//            children of level1 node j1 = g-64)
#define NODES   17472
#define BATCHN  4096
#define NGRP16  1088
#define NL1     1024
#define THREADS 256

// ---- gfx1250 async global->LDS path (ASYNCcnt), guarded so we always compile
#if defined(__has_builtin)
#  if __has_builtin(__builtin_amdgcn_global_load_async_to_lds_b128) && \
      __has_builtin(__builtin_amdgcn_s_wait_asynccnt)
#    define USE_ASYNC_LDS 1
#  endif
#endif

#ifdef USE_ASYNC_LDS
// Toolchain signature (from round-1 diagnostic): param0 is
// 'int __vector(4) __device__ *' i.e. addrspace(1) pointer to a gcc-vector
// int4; param1 is the LDS-side pointer (addrspace(3)).
typedef int v4i_t __attribute__((vector_size(4 * sizeof(int))));
typedef __attribute__((address_space(1))) v4i_t* g_v4i_ptr;
typedef __attribute__((address_space(3))) v4i_t* l_v4i_ptr;
#endif

__device__ __forceinline__ void copy16_g2l(const float* g, float* l) {
#ifdef USE_ASYNC_LDS
  __builtin_amdgcn_global_load_async_to_lds_b128(
      (g_v4i_ptr)(void*)g, (l_v4i_ptr)(void*)l,
      /*offset=*/0, /*cpol=*/0);
#else
  *(float4*)l = *(const float4*)g;
#endif
}

__device__ __forceinline__ float wave_sum32(float v) {
#pragma unroll
  for (int m = 16; m >= 1; m >>= 1) v += __shfl_xor(v, m, 32);
  return v;
}
__device__ __forceinline__ float wave_max32(float v) {
#pragma unroll
  for (int m = 16; m >= 1; m >>= 1) v = fmaxf(v, __shfl_xor(v, m, 32));
  return v;
}

__global__ __launch_bounds__(THREADS) void hsm_fwd(
    const float* __restrict__ pred,   // [B, N]
    const float* __restrict__ tgt,    // [B, N]
    float* __restrict__ probs,        // [B, N] (= d_out + 1)
    float* __restrict__ part) {       // [B] per-sample sum(logp*target)
  __shared__ __align__(16) float s_pred[NODES];   // whole row, 68.25 KB
  __shared__ float s_logZ[NGRP16];                // per 16-group logZ
  __shared__ float s_L1[NL1];                     // cumulative level1 logp
  __shared__ float s_red[8];
  __shared__ float s_d0;

  const int tid = threadIdx.x;
  const int b   = blockIdx.x;
  const float* rowp = pred + (size_t)b * NODES;
  const float* trow = tgt  + (size_t)b * NODES;

  // ---- Phase 1: DMA the sample row into LDS (17472 floats = 4368 x b128)
  for (int c = tid; c < NODES / 4; c += THREADS)
    copy16_g2l(rowp + 4 * c, &s_pred[4 * c]);
  // overlap: prefetch the target row (one lane per 128B cacheline)
  for (int off = tid * 32; off < NODES; off += THREADS * 32)
    __builtin_prefetch(trow + off, 0, 3);
#ifdef USE_ASYNC_LDS
  __builtin_amdgcn_s_wait_asynccnt(0);
#endif
  __syncthreads();

  // ---- Phase 2a: root group logZ over 64 level0 nodes (wave 0 only)
  if (tid < 32) {
    float v0 = s_pred[tid], v1 = s_pred[tid + 32];
    float m  = wave_max32(fmaxf(v0, v1));
    float s  = wave_sum32(expf(v0 - m) + expf(v1 - m));
    if (tid == 0) s_d0 = m + logf(s);
  }

  // ---- Phase 2b: 1088 contiguous 16-elem groups; 16 lanes per group
  {
    const int sub  = tid >> 4;   // 0..15 concurrent groups
    const int lane = tid & 15;
    for (int g = sub; g < NGRP16; g += 16) {
      float v = s_pred[64 + 16 * g + lane];
      float m = v;
      m = fmaxf(m, __shfl_xor(m, 1, 32));
      m = fmaxf(m, __shfl_xor(m, 2, 32));
      m = fmaxf(m, __shfl_xor(m, 4, 32));
      m = fmaxf(m, __shfl_xor(m, 8, 32));
      float e = expf(v - m);
      e += __shfl_xor(e, 1, 32);
      e += __shfl_xor(e, 2, 32);
      e += __shfl_xor(e, 4, 32);
      e += __shfl_xor(e, 8, 32);
      if (lane == 0) s_logZ[g] = m + logf(e);
    }
  }
  __syncthreads();

  const float d0 = s_d0;

  // ---- Phase 3: cumulative level1 log-probs: L1[j1] = L0[p] + local1[j1]
  for (int j1 = tid; j1 < NL1; j1 += THREADS) {
    const int p = j1 >> 4;                       // level0 parent
    s_L1[j1] = (s_pred[p] - d0) + s_pred[64 + j1] - s_logZ[p];
  }
  __syncthreads();

  // ---- Phase 4: streaming output pass (coalesced), loss dot-product
  float acc = 0.f;
  float* prow = probs + (size_t)b * NODES;
  for (int i = tid; i < NODES; i += THREADS) {
    float lp;
    if (i >= 1088) {                             // leaves
      const int t = i - 1088, j1 = t >> 4;
      lp = s_L1[j1] + s_pred[i] - s_logZ[64 + j1];
    } else if (i >= 64) {                        // level1
      lp = s_L1[i - 64];
    } else {                                     // level0
      lp = s_pred[i] - d0;
    }
    prow[i] = expf(lp);
    acc += lp * trow[i];
  }

  // deterministic block reduction
  acc = wave_sum32(acc);
  if ((tid & 31) == 0) s_red[tid >> 5] = acc;
  __syncthreads();
  if (tid == 0) {
    float t = 0.f;
#pragma unroll
    for (int w = 0; w < 8; ++w) t += s_red[w];
    part[b] = t;
  }
}

__global__ __launch_bounds__(THREADS) void hsm_loss(
    const float* __restrict__ part, float* __restrict__ out) {
  __shared__ float s_red[8];
  const int tid = threadIdx.x;
  float a = 0.f;
  for (int i = tid; i < BATCHN; i += THREADS) a += part[i];  // fixed order
  a = wave_sum32(a);
  if ((tid & 31) == 0) s_red[tid >> 5] = a;
  __syncthreads();
  if (tid == 0) {
    float t = 0.f;
#pragma unroll
    for (int w = 0; w < 8; ++w) t += s_red[w];
    out[0] = -t / (float)BATCHN;
  }
}

extern "C" void kernel_launch(void* const* d_in, const int* in_sizes, int n_in,
                              void* d_out, int out_size, void* d_ws, size_t ws_size,
                              hipStream_t stream) {
  (void)in_sizes; (void)n_in; (void)out_size; (void)ws_size;
  const float* pred = (const float*)d_in[0];   // [4096, 17472] f32
  const float* tgt  = (const float*)d_in[1];   // [4096, 17472] f32
  float* out   = (float*)d_out;                // [0]=loss, [1..]=probs
  float* probs = out + 1;
  float* part  = (float*)d_ws;                 // 4096 floats of scratch

  hsm_fwd<<<BATCHN, THREADS, 0, stream>>>(pred, tgt, probs, part);
  hsm_loss<<<1, THREADS, 0, stream>>>(part, out);
}